// CellPathwayAttentionAggregator_72782515798454
// MI455X (gfx1250) — compile-verified
//
#include <hip/hip_runtime.h>
#include <hip/hip_bf16.h>
#include <math.h>

#define NUM_SETS  2048
#define NUM_GENES 8192
#define BATCH     1024

typedef __attribute__((ext_vector_type(2))) float v2f;
typedef __attribute__((ext_vector_type(8))) float v8f;

// ---------------------------------------------------------------------------
// Kernel 1: segment start offsets. segment_ids is sorted (repeat(arange,sizes)),
// every segment is non-empty (sizes >= 8), so every start[s] gets written.
// ---------------------------------------------------------------------------
__global__ void seg_bounds_kernel(const int* __restrict__ seg, int total,
                                  int* __restrict__ start) {
    int j = blockIdx.x * blockDim.x + threadIdx.x;
    if (j >= total) return;
    if (j == 0) {
        start[seg[0]]   = 0;
        start[NUM_SETS] = total;
    } else if (seg[j] != seg[j - 1]) {
        start[seg[j]] = j;
    }
}

// ---------------------------------------------------------------------------
// Kernel 2: per-segment softmax weights, packed with gather index and the
// segment's row-within-16-tile:  einfo[j] = { w_j , bits(idx_j | (s&15)<<13) }.
// One 128-thread block per segment (max size 120).
// ---------------------------------------------------------------------------
__global__ void seg_softmax_kernel(const float* __restrict__ logits,
                                   const int*   __restrict__ flat_idx,
                                   const int*   __restrict__ start,
                                   float2*      __restrict__ einfo) {
    __shared__ float red[128];
    const int s  = blockIdx.x;
    const int j0 = start[s];
    const int n  = start[s + 1] - j0;
    const int t  = threadIdx.x;

    float x = (t < n) ? logits[j0 + t] : -INFINITY;
    red[t] = x; __syncthreads();
    for (int off = 64; off > 0; off >>= 1) {
        if (t < off) red[t] = fmaxf(red[t], red[t + off]);
        __syncthreads();
    }
    const float mx = red[0]; __syncthreads();

    float e = (t < n) ? expf(x - mx) : 0.0f;
    red[t] = e; __syncthreads();
    for (int off = 64; off > 0; off >>= 1) {
        if (t < off) red[t] += red[t + off];
        __syncthreads();
    }
    const float denom = red[0];

    if (t < n) {
        float2 v;
        v.x = e / denom;
        v.y = __int_as_float(flat_idx[j0 + t] | ((s & 15) << 13));
        einfo[j0 + t] = v;
    }
}

// ---------------------------------------------------------------------------
// Kernel 3: F(1024 x 8192) -> FT(8192 x 1024) so gathered geneset columns are
// contiguous over batch. 32x32 LDS tile, both sides coalesced.
// ---------------------------------------------------------------------------
__global__ void transpose_kernel(const float* __restrict__ F,
                                 float* __restrict__ FT) {
    __shared__ float tile[32][33];
    const int g0 = blockIdx.x * 32;
    const int b0 = blockIdx.y * 32;
    const int tx = threadIdx.x;   // 0..31
    const int ty = threadIdx.y;   // 0..7
    #pragma unroll
    for (int r = 0; r < 32; r += 8)
        tile[ty + r][tx] = F[(size_t)(b0 + ty + r) * NUM_GENES + g0 + tx];
    __syncthreads();
    #pragma unroll
    for (int r = 0; r < 32; r += 8)
        FT[(size_t)(g0 + ty + r) * BATCH + b0 + tx] = tile[tx][ty + r];
}

// ---------------------------------------------------------------------------
// Kernel 4: WMMA SpMM.
//   Grid: x = 16 groups of 128 segments, y = 64 tiles of 16 batch rows.
//   Block: 256 threads = 8 waves; wave w owns segments [x*128+16w, +16).
//   D(16 seg x 16 batch) = sum_K A(16x4)*B(4x16) via V_WMMA_F32_16X16X4_F32.
//   A[m,k] = w_k iff m == seg_k mod 16 (one nonzero per K column);
//   B[k,n] = FT[g_k, b0+n] (64B coalesced per half-wave).
//   K loop unrolled x2 with two accumulators: 4 gathers in flight per
//   iteration, halved WMMA D->C dependency chain.
//   Loop bounds are readfirstlane'd to SGPRs: scalar branch, EXEC stays ~0
//   churn and all-ones for WMMA. Tail = clamped addresses + w=0 selects.
// ISA layouts (CDNA5 ISA):
//   A 16x4 f32: lanes 0-15 hold K=0 (V0), K=1 (V1); lanes 16-31 hold K=2,3.
//   B mirrored: V0 = K row (2*half), V1 = K row (2*half+1).
//   D 16x16 f32: V_v holds (M = v + 8*half, N = lane&15).
// ---------------------------------------------------------------------------
#define TO_PAD 132   // LDS row stride in floats (16B-aligned, bank-spread)

template <int USE_FT>
__global__ void __launch_bounds__(256)
spmm_wmma_kernel(const float*  __restrict__ feat,   // FT if USE_FT else F
                 const float2* __restrict__ einfo,
                 const int*    __restrict__ start,
                 float*        __restrict__ out) {
    __shared__ float tileout[16 * TO_PAD];   // [batch 16][seg 128 (+pad)]

    const int tid  = threadIdx.x;
    const int wave = tid >> 5;
    const int lane = tid & 31;
    const int half = lane >> 4;           // entry-pair owned by this half-wave
    const int n    = lane & 15;           // batch column within tile
    const int s0   = blockIdx.x * 128 + wave * 16;   // 16-aligned seg base
    const int b    = blockIdx.y * 16 + n;

    const int jb = __builtin_amdgcn_readfirstlane(start[s0]);
    const int je = __builtin_amdgcn_readfirstlane(start[s0 + 16]);

    v8f c0 = {}, c1 = {};
    for (int j = jb; j < je; j += 8) {
        __builtin_prefetch(&einfo[j + 64], 0, 1);   // global_prefetch_b8

        #pragma unroll
        for (int u = 0; u < 2; u++) {
            const int j0  = j + 4 * u + 2 * half;
            const int j1  = j0 + 1;
            const int j0c = (j0 < je) ? j0 : je - 1;   // clamp, no divergence
            const int j1c = (j1 < je) ? j1 : je - 1;

            const float2 e0 = einfo[j0c];
            const float2 e1 = einfo[j1c];
            const int p0 = __float_as_int(e0.y);
            const int p1 = __float_as_int(e1.y);
            const int g0 = p0 & 8191, m0 = p0 >> 13;
            const int g1 = p1 & 8191, m1 = p1 >> 13;
            const float w0 = (j0 < je) ? e0.x : 0.0f;  // w=0 kills pad columns
            const float w1 = (j1 < je) ? e1.x : 0.0f;

            v2f a, bm;
            a.x = (m0 == n) ? w0 : 0.0f;
            a.y = (m1 == n) ? w1 : 0.0f;
            if (USE_FT) {   // 32-bit element offsets -> saddr+voffset form
                bm.x = feat[(unsigned)g0 * BATCH + (unsigned)b];
                bm.y = feat[(unsigned)g1 * BATCH + (unsigned)b];
            } else {
                bm.x = feat[(size_t)b * NUM_GENES + g0];
                bm.y = feat[(size_t)b * NUM_GENES + g1];
            }

            if (u == 0)
                c0 = __builtin_amdgcn_wmma_f32_16x16x4_f32(
                        false, a, false, bm, (short)0, c0, false, false);
            else
                c1 = __builtin_amdgcn_wmma_f32_16x16x4_f32(
                        false, a, false, bm, (short)0, c1, false, false);
        }
    }
    const v8f c = c0 + c1;

    // Stage D into LDS: lane holds rows m = v + 8*half, column n.
    // Consecutive v -> consecutive floats => 2x ds_store_b128 per lane.
    #pragma unroll
    for (int v = 0; v < 8; v++) {
        const int m = v + 8 * half;
        tileout[n * TO_PAD + wave * 16 + m] = c[v];
    }
    __syncthreads();

    // Coalesced write-out: 512 float4 over 256 threads, 2 each.
    const int bbase = blockIdx.y * 16;
    const int sbase = blockIdx.x * 128;
    #pragma unroll
    for (int r = 0; r < 2; r++) {
        const int i  = tid + r * 256;
        const int bb = i >> 5;          // 32 float4 per batch row
        const int s4 = (i & 31) << 2;
        const float4 v = *(const float4*)&tileout[bb * TO_PAD + s4];
        *(float4*)&out[(size_t)(bbase + bb) * NUM_SETS + sbase + s4] = v;
    }
}

// ---------------------------------------------------------------------------
extern "C" void kernel_launch(void* const* d_in, const int* in_sizes, int n_in,
                              void* d_out, int out_size, void* d_ws, size_t ws_size,
                              hipStream_t stream) {
    const float* feat     = (const float*)d_in[0];   // (1024, 8192) f32
    const float* logits   = (const float*)d_in[1];   // (total,)     f32
    const int*   flat_idx = (const int*)  d_in[2];   // (total,)     i32
    const int*   seg      = (const int*)  d_in[3];   // (total,)     i32
    float*       out      = (float*)d_out;           // (1024, 2048) f32
    const int    total    = in_sizes[1];

    // Workspace layout: [start: 16KB][einfo: total*8B][FT: 32MB, 256B-aligned]
    char*   ws    = (char*)d_ws;
    int*    start = (int*)ws;
    float2* einfo = (float2*)(ws + 16384);
    size_t  ft_off = (16384 + (size_t)total * 8 + 255) & ~(size_t)255;
    float*  FT    = (float*)(ws + ft_off);
    const size_t need_ft = ft_off + (size_t)NUM_GENES * BATCH * sizeof(float);
    const int use_ft = (ws_size >= need_ft) ? 1 : 0;

    seg_bounds_kernel<<<(total + 255) / 256, 256, 0, stream>>>(seg, total, start);
    seg_softmax_kernel<<<NUM_SETS, 128, 0, stream>>>(logits, flat_idx, start, einfo);

    const dim3 grid(NUM_SETS / 128, BATCH / 16);
    if (use_ft) {
        transpose_kernel<<<dim3(NUM_GENES / 32, BATCH / 32), dim3(32, 8), 0, stream>>>(feat, FT);
        spmm_wmma_kernel<1><<<grid, 256, 0, stream>>>(FT, einfo, start, out);
    } else {
        spmm_wmma_kernel<0><<<grid, 256, 0, stream>>>(feat, einfo, start, out);
    }
}